// AdaLayerNormZeroSingle_36112085025538
// MI455X (gfx1250) — compile-verified
//
#include <hip/hip_runtime.h>
#include <hip/hip_bf16.h>
#include <math.h>

typedef __attribute__((ext_vector_type(2))) float v2f;
typedef __attribute__((ext_vector_type(8))) float v8f;

#define DIM  3072
#define SEQ  8192
#define ODIM (3 * DIM) /* 9216 */

// ---------------------------------------------------------------------------
// Kernel A: e = silu(emb) @ W^T + b  via V_WMMA_F32_16X16X4_F32.
// Block = 128 threads (4 waves). Each wave computes 16 output columns.
// A-matrix (16x4 f32): lane layout vgpr=k%2, half=k/2 -> lane in half h needs
// {s[k+2h], s[k+2h+1]}; we broadcast the silu vector to all 16 rows so D's
// rows are all equal and row 0 (acc[0], lanes 0-15) is the GEMV result.
// B-matrix (4x16 f32): same (vgpr=k%2, half=k/2) striping -> lane (n,h) needs
// {W[n0+n, k+2h], W[n0+n, k+2h+1]} = one contiguous global_load_b64.
// Results are deinterleaved: g%3==0 -> shift[g/3], 1 -> scale[g/3],
// 2 -> gate[g/3] (written straight into the tail of d_out).
// ---------------------------------------------------------------------------
__global__ __launch_bounds__(128) void gemv_silu_wmma(
    const float* __restrict__ emb, const float* __restrict__ W,
    const float* __restrict__ bias, float* __restrict__ shift,
    float* __restrict__ scale, float* __restrict__ gate)
{
    __shared__ float sS[DIM]; // silu(emb), 12 KB of the 320 KB WGP LDS

    const int tid = threadIdx.x;
    for (int i = tid; i < DIM; i += 128) {
        float v = emb[i];
        sS[i] = v / (1.0f + expf(-v)); // silu
    }
    __syncthreads();

    const int lane = tid & 31;
    const int wave = tid >> 5;
    const int h    = lane >> 4;  // which 16-lane half
    const int n    = lane & 15;  // output column within tile
    const int n0   = blockIdx.x * 64 + wave * 16;

    const float* wrow = W + (size_t)(n0 + n) * DIM;

    v8f acc = {};
    for (int k = 0; k < DIM; k += 4) {
        const int kk = k + 2 * h;
        if ((k & 63) == 0) {
            // stream-ahead hint on the W row -> global_prefetch_b8
            __builtin_prefetch(wrow + k + 128, 0, 0);
        }
        v2f a  = *reinterpret_cast<const v2f*>(&sS[kk]);   // ds_load_b64
        v2f bm = *reinterpret_cast<const v2f*>(wrow + kk); // global_load_b64
        acc = __builtin_amdgcn_wmma_f32_16x16x4_f32(
            /*neg_a=*/false, a, /*neg_b=*/false, bm,
            /*c_mod=*/(short)0, acc, /*reuse_a=*/false, /*reuse_b=*/false);
    }

    if (lane < 16) {
        const int g = n0 + lane;          // global output index in [0, 9216)
        const float e = acc[0] + bias[g]; // D[0][lane] + bias
        const int comp = g % 3;
        const int idx  = g / 3;
        if (comp == 0)      shift[idx] = e;
        else if (comp == 1) scale[idx] = e;
        else                gate[idx]  = e;
    }
}

// ---------------------------------------------------------------------------
// Kernel B: LayerNorm(x) * scale + shift, one row (3072 f32) per 256-thread
// block. x is loaded once into registers (3x float4 per thread, b128 loads),
// statistics via LDS tree reduction, apply from registers, b128 stores.
// ---------------------------------------------------------------------------
__global__ __launch_bounds__(256) void ln_modulate(
    const float* __restrict__ x, const float* __restrict__ shift,
    const float* __restrict__ scale, float* __restrict__ out)
{
    const int row = blockIdx.x;
    const int tid = threadIdx.x;

    const float4* xr   = reinterpret_cast<const float4*>(x + (size_t)row * DIM);
    float4*       orow = reinterpret_cast<float4*>(out + (size_t)row * DIM);

    float4 v[3];
    float sum = 0.0f, sq = 0.0f;
#pragma unroll
    for (int j = 0; j < 3; ++j) {
        v[j] = xr[tid + j * 256];
        sum += v[j].x + v[j].y + v[j].z + v[j].w;
        sq  += v[j].x * v[j].x + v[j].y * v[j].y +
               v[j].z * v[j].z + v[j].w * v[j].w;
    }

    __shared__ float sSum[256];
    __shared__ float sSq[256];
    sSum[tid] = sum;
    sSq[tid]  = sq;
    __syncthreads();
#pragma unroll
    for (int off = 128; off > 0; off >>= 1) {
        if (tid < off) {
            sSum[tid] += sSum[tid + off];
            sSq[tid]  += sSq[tid + off];
        }
        __syncthreads();
    }

    const float mean = sSum[0] * (1.0f / (float)DIM);
    const float var  = sSq[0] * (1.0f / (float)DIM) - mean * mean;
    const float rs   = rsqrtf(var + 1e-6f);

    const float4* sh4 = reinterpret_cast<const float4*>(shift);
    const float4* sc4 = reinterpret_cast<const float4*>(scale);
#pragma unroll
    for (int j = 0; j < 3; ++j) {
        const int c4 = tid + j * 256;
        const float4 sh = sh4[c4];
        const float4 sc = sc4[c4];
        float4 o;
        o.x = (v[j].x - mean) * rs * sc.x + sh.x;
        o.y = (v[j].y - mean) * rs * sc.y + sh.y;
        o.z = (v[j].z - mean) * rs * sc.z + sh.z;
        o.w = (v[j].w - mean) * rs * sc.w + sh.w;
        orow[c4] = o;
    }
}

extern "C" void kernel_launch(void* const* d_in, const int* in_sizes, int n_in,
                              void* d_out, int out_size, void* d_ws, size_t ws_size,
                              hipStream_t stream)
{
    // setup_inputs order: x [1,8192,3072], emb [1,3072], W [9216,3072], b [9216]
    const float* x    = (const float*)d_in[0];
    const float* emb  = (const float*)d_in[1];
    const float* W    = (const float*)d_in[2];
    const float* bias = (const float*)d_in[3];

    float* out = (float*)d_out;                   // [8192*3072] out, then [3072] gate
    float* ws  = (float*)d_ws;
    float* shift = ws;                            // 3072 f32
    float* scale = ws + DIM;                      // 3072 f32
    float* gate  = out + (size_t)SEQ * DIM;       // tail of d_out

    // Stage 1: modulation GEMV (WMMA), writes shift/scale (ws) + gate (d_out)
    gemv_silu_wmma<<<ODIM / 64, 128, 0, stream>>>(emb, W, bias, shift, scale, gate);

    // Stage 2: fused LayerNorm + modulate, one row per block
    ln_modulate<<<SEQ, 256, 0, stream>>>(x, shift, scale, out);
}